// KinematicModel_60352880443724
// MI455X (gfx1250) — compile-verified
//
#include <hip/hip_runtime.h>

// Problem constants (match reference).
#define V_N  200000
#define S_N  300
#define NJ   24
#define KN   32

typedef __attribute__((ext_vector_type(2))) float v2f;
typedef __attribute__((ext_vector_type(8))) float v8f;

// ---------------------------------------------------------------------------
// Kernel 0: zero the accumulators (joint positions in ws, keypoints in d_out).
// ---------------------------------------------------------------------------
__global__ void k_init(float* __restrict__ Jws, float* __restrict__ kp) {
    int t = blockIdx.x * blockDim.x + threadIdx.x;
    if (t < NJ * 3) Jws[t] = 0.0f;
    if (t < KN * 3) kp[t]  = 0.0f;
}

// ---------------------------------------------------------------------------
// Kernel 1: verts = template + mesh_shape_basis[V*3, 300] . shape[300]
// Streams the 720 MB basis once; one thread per output row, float4 loads,
// shape vector staged in LDS. Bandwidth-bound -> this kernel sets total time.
// ---------------------------------------------------------------------------
__global__ void k_shape(const float* __restrict__ basis,
                        const float* __restrict__ tmpl,
                        const float* __restrict__ shape,
                        float* __restrict__ verts) {
    __shared__ float sh[S_N];
    for (int i = threadIdx.x; i < S_N; i += blockDim.x) sh[i] = shape[i];
    __syncthreads();

    int row = blockIdx.x * blockDim.x + threadIdx.x;
    if (row >= V_N * 3) return;

    const float* b = basis + (size_t)row * S_N;     // 1200 B, 16B-aligned
    __builtin_prefetch(b, 0, 0);                    // global_prefetch_b8
    float acc = 0.0f;
#pragma unroll 5
    for (int s = 0; s < S_N; s += 4) {
        float4 q = *(const float4*)(b + s);
        acc += q.x * sh[s] + q.y * sh[s + 1] + q.z * sh[s + 2] + q.w * sh[s + 3];
    }
    verts[row] = tmpl[row] + acc;
}

// ---------------------------------------------------------------------------
// Kernel 2/5: out3[row] += W[row, :] . x3[:, c]   (row = blockIdx.y)
// Grid-stride partial sums, LDS tree reduction, global f32 atomics.
// Used for J = Jr @ verts (24 rows) and keypoints = Jre @ out_v (32 rows).
// ---------------------------------------------------------------------------
__global__ void k_reduce(const float* __restrict__ W,
                         const float* __restrict__ x3,
                         float* __restrict__ out3) {
    const int row = blockIdx.y;
    const float* w = W + (size_t)row * V_N;
    float sx = 0.f, sy = 0.f, sz = 0.f;
    for (int v = blockIdx.x * blockDim.x + threadIdx.x; v < V_N;
         v += gridDim.x * blockDim.x) {
        float wv = w[v];
        sx += wv * x3[3 * v + 0];
        sy += wv * x3[3 * v + 1];
        sz += wv * x3[3 * v + 2];
    }
    __shared__ float r0[256], r1[256], r2[256];
    r0[threadIdx.x] = sx; r1[threadIdx.x] = sy; r2[threadIdx.x] = sz;
    __syncthreads();
    for (int off = 128; off > 0; off >>= 1) {
        if (threadIdx.x < off) {
            r0[threadIdx.x] += r0[threadIdx.x + off];
            r1[threadIdx.x] += r1[threadIdx.x + off];
            r2[threadIdx.x] += r2[threadIdx.x + off];
        }
        __syncthreads();
    }
    if (threadIdx.x == 0) {
        atomicAdd(&out3[3 * row + 0], r0[0]);
        atomicAdd(&out3[3 * row + 1], r1[0]);
        atomicAdd(&out3[3 * row + 2], r2[0]);
    }
}

// ---------------------------------------------------------------------------
// Kernel 3: Rodrigues + sequential kinematic chain + rest-pose correction.
// One wave. Lanes 0..23 build rotation matrices in parallel; lane 0 walks the
// 24-joint parent chain (3x4 compositions, implicit [0 0 0 1] bottom row);
// lanes 0..23 pack G' into a [24][16] row-major B-matrix (cols 12..15 = 0).
// ---------------------------------------------------------------------------
__global__ void k_chain(const float* __restrict__ pose,
                        const float* __restrict__ J,
                        const int*   __restrict__ parents,
                        float* __restrict__ Gp) {
    __shared__ float Rm[NJ][9];
    __shared__ float Gl[NJ][12];
    const int j = threadIdx.x;
    if (j < NJ) {
        float rx = pose[3 * j], ry = pose[3 * j + 1], rz = pose[3 * j + 2];
        float th = sqrtf(rx * rx + ry * ry + rz * rz) + 1e-8f;
        float inv = 1.0f / th;
        float ux = rx * inv, uy = ry * inv, uz = rz * inv;
        float c = cosf(th), s = sinf(th), o = 1.0f - c;
        Rm[j][0] = c + o * ux * ux;     Rm[j][1] = o * ux * uy - s * uz; Rm[j][2] = o * ux * uz + s * uy;
        Rm[j][3] = o * uy * ux + s * uz; Rm[j][4] = c + o * uy * uy;     Rm[j][5] = o * uy * uz - s * ux;
        Rm[j][6] = o * uz * ux - s * uy; Rm[j][7] = o * uz * uy + s * ux; Rm[j][8] = c + o * uz * uz;
    }
    __syncthreads();
    if (threadIdx.x == 0) {
        for (int r = 0; r < 3; ++r) {
            for (int cc = 0; cc < 3; ++cc) Gl[0][r * 4 + cc] = Rm[0][r * 3 + cc];
            Gl[0][r * 4 + 3] = J[r];
        }
        for (int i = 1; i < NJ; ++i) {
            int p = parents[i];
            float tx = J[3 * i + 0] - J[3 * p + 0];
            float ty = J[3 * i + 1] - J[3 * p + 1];
            float tz = J[3 * i + 2] - J[3 * p + 2];
            for (int r = 0; r < 3; ++r) {
                float g0 = Gl[p][r * 4 + 0], g1 = Gl[p][r * 4 + 1];
                float g2 = Gl[p][r * 4 + 2], g3 = Gl[p][r * 4 + 3];
                for (int cc = 0; cc < 3; ++cc)
                    Gl[i][r * 4 + cc] = g0 * Rm[i][cc] + g1 * Rm[i][3 + cc] + g2 * Rm[i][6 + cc];
                Gl[i][r * 4 + 3] = g0 * tx + g1 * ty + g2 * tz + g3;
            }
        }
    }
    __syncthreads();
    if (j < NJ) {
        float jx = J[3 * j], jy = J[3 * j + 1], jz = J[3 * j + 2];
        for (int r = 0; r < 3; ++r) {
            float corr = Gl[j][r * 4 + 0] * jx + Gl[j][r * 4 + 1] * jy + Gl[j][r * 4 + 2] * jz;
            Gp[j * 16 + r * 4 + 0] = Gl[j][r * 4 + 0];
            Gp[j * 16 + r * 4 + 1] = Gl[j][r * 4 + 1];
            Gp[j * 16 + r * 4 + 2] = Gl[j][r * 4 + 2];
            Gp[j * 16 + r * 4 + 3] = Gl[j][r * 4 + 3] - corr;
        }
        Gp[j * 16 + 12] = 0.f; Gp[j * 16 + 13] = 0.f;
        Gp[j * 16 + 14] = 0.f; Gp[j * 16 + 15] = 0.f;
    }
}

// ---------------------------------------------------------------------------
// Kernel 4: WMMA skinning. T = sw[V,24] @ Gp[24,16] as a GEMM:
//   M = 16 vertices per wave, N = 16 (12 used: flattened 3x4 T), K = 24 in
//   six V_WMMA_F32_16X16X4_F32 steps. Accumulator tile spilled through LDS so
//   lanes 0..15 can apply their vertex's T to (vx,vy,vz,1), scale, origin.
// A-frag layout (32-bit A 16x4): lanes 0-15 hold K=0(K=1 in .y), lanes 16-31
// hold K=2,3 for the same M rows. B-frag mirrors it over N lanes.
// verts are read from and out_v written to the same buffer (d_out), each lane
// only touches its own vertex, read-before-write.
// ---------------------------------------------------------------------------
__global__ void k_skin(const float* __restrict__ sw,
                       const float* __restrict__ Gp,
                       float* __restrict__ vio,          // in: verts, out: out_v
                       const float* __restrict__ scale_p,
                       const float* __restrict__ origin) {
    __shared__ float GpL[NJ * 16];        // 1.5 KB
    __shared__ float tile[8 * 256];       // 8 waves x 16x16 f32 accum = 8 KB
    for (int i = threadIdx.x; i < NJ * 16; i += blockDim.x) GpL[i] = Gp[i];
    __syncthreads();

    const int lane = threadIdx.x & 31;
    const int wave = threadIdx.x >> 5;
    const int v0   = (blockIdx.x * 8 + wave) * 16;   // V_N % 16 == 0
    const bool active = (v0 < V_N);                  // wave-uniform
    float* wtile = tile + wave * 256;

    if (active) {
        const int m     = lane & 15;
        const int khalf = (lane >> 4) << 1;          // 0 or 2
        v8f acc = {0.f, 0.f, 0.f, 0.f, 0.f, 0.f, 0.f, 0.f};
#pragma unroll
        for (int kc = 0; kc < NJ / 4; ++kc) {
            const int kb = kc * 4 + khalf;
            const float* ap = sw + (size_t)(v0 + m) * NJ + kb;   // 8B aligned
            v2f a, b;
            a.x = ap[0];
            a.y = ap[1];
            b.x = GpL[kb * 16 + m];
            b.y = GpL[(kb + 1) * 16 + m];
            acc = __builtin_amdgcn_wmma_f32_16x16x4_f32(
                /*neg_a=*/false, a, /*neg_b=*/false, b,
                /*c_mod=*/(short)0, acc, /*reuse_a=*/false, /*reuse_b=*/false);
        }
        const int mbase = (lane >> 4) << 3;          // 0 or 8
#pragma unroll
        for (int r = 0; r < 8; ++r)
            wtile[(mbase + r) * 16 + m] = acc[r];
    }
    __syncthreads();

    if (active && lane < 16) {
        const int v = v0 + lane;
        const float* T = wtile + lane * 16;          // flattened 3x4 transform
        float vx = vio[3 * v + 0], vy = vio[3 * v + 1], vz = vio[3 * v + 2];
        float sc = scale_p[0];
        float ox = T[0] * vx + T[1] * vy + T[2]  * vz + T[3];
        float oy = T[4] * vx + T[5] * vy + T[6]  * vz + T[7];
        float oz = T[8] * vx + T[9] * vy + T[10] * vz + T[11];
        vio[3 * v + 0] = ox * sc - origin[0];
        vio[3 * v + 1] = oy * sc - origin[1];
        vio[3 * v + 2] = oz * sc - origin[2];
    }
}

// ---------------------------------------------------------------------------
// Host-side launch. Inputs (setup_inputs order):
// 0 pose[24,3] 1 shape[300] 2 mesh_template[V,3] 3 mesh_shape_basis[V,3,300]
// 4 J_regressor[24,V] 5 skinning_weights[V,24] 6 J_regressor_ext[32,V]
// 7 scale[1] 8 coord_origin[3] 9 parents[24] (int)
// d_out = out_v[V,3] ++ keypoints[32,3]  (600096 f32)
// ws: [0..71] joint accumulator, [128..511] packed G (24x16) -> 2 KB used.
// ---------------------------------------------------------------------------
extern "C" void kernel_launch(void* const* d_in, const int* in_sizes, int n_in,
                              void* d_out, int out_size, void* d_ws, size_t ws_size,
                              hipStream_t stream) {
    const float* pose    = (const float*)d_in[0];
    const float* shape   = (const float*)d_in[1];
    const float* tmpl    = (const float*)d_in[2];
    const float* basis   = (const float*)d_in[3];
    const float* Jr      = (const float*)d_in[4];
    const float* sw      = (const float*)d_in[5];
    const float* Jre     = (const float*)d_in[6];
    const float* scale   = (const float*)d_in[7];
    const float* origin  = (const float*)d_in[8];
    const int*   parents = (const int*)d_in[9];

    float* outv = (float*)d_out;           // [V*3], also holds verts temporarily
    float* kp   = outv + V_N * 3;          // [32*3]
    float* Jws  = (float*)d_ws;            // 72 floats
    float* Gp   = Jws + 128;               // 384 floats, 16B aligned

    k_init<<<1, 128, 0, stream>>>(Jws, kp);
    k_shape<<<(V_N * 3 + 255) / 256, 256, 0, stream>>>(basis, tmpl, shape, outv);
    k_reduce<<<dim3(64, NJ), 256, 0, stream>>>(Jr, outv, Jws);
    k_chain<<<1, 32, 0, stream>>>(pose, Jws, parents, Gp);
    k_skin<<<(V_N + 127) / 128, 256, 0, stream>>>(sw, Gp, outv, scale, origin);
    k_reduce<<<dim3(64, KN), 256, 0, stream>>>(Jre, outv, kp);
}